// My_MSE_loss_21380347200134
// MI455X (gfx1250) — compile-verified
//
#include <hip/hip_runtime.h>

typedef __attribute__((ext_vector_type(2))) float v2f;
typedef __attribute__((ext_vector_type(8))) float v8f;

constexpr int kThreads = 256;   // 8 wave32s per block
constexpr int kBlocks  = 2048;  // 2048 partials -> d_ws

__device__ __forceinline__ float wave32_reduce(float v) {
    // wave32: reduce across 32 lanes
    #pragma unroll
    for (int o = 16; o > 0; o >>= 1) v += __shfl_down(v, o, 32);
    return v;
}

__device__ __forceinline__ float huber1(float d) {
    float ad = fabsf(d);
    float q  = 0.5f * d * d;
    return (ad <= 1.0f) ? q : ad;   // delta = 1 Huber, matches reference
}

// Stage 1: streaming pass, HBM-bandwidth bound. float4 loads -> global_load_b128.
__global__ void huber_partial_kernel(const float* __restrict__ x,
                                     const float* __restrict__ y,
                                     float* __restrict__ partial, int n) {
    const int tid    = blockIdx.x * blockDim.x + threadIdx.x;
    const int stride = gridDim.x * blockDim.x;
    const int n4     = n >> 2;

    const float4* __restrict__ x4 = (const float4*)x;
    const float4* __restrict__ y4 = (const float4*)y;

    float acc = 0.0f;
    for (int i = tid; i < n4; i += stride) {
        float4 a = x4[i];
        float4 b = y4[i];
        acc += huber1(a.x - b.x);
        acc += huber1(a.y - b.y);
        acc += huber1(a.z - b.z);
        acc += huber1(a.w - b.w);
    }
    // scalar tail (N % 4), generically safe
    for (int i = (n4 << 2) + tid; i < n; i += stride) {
        acc += huber1(x[i] - y[i]);
    }

    // wave32 reduce, then cross-wave via LDS
    acc = wave32_reduce(acc);
    __shared__ float lds[kThreads / 32];
    const int lane = threadIdx.x & 31;
    const int wid  = threadIdx.x >> 5;
    if (lane == 0) lds[wid] = acc;
    __syncthreads();
    if (threadIdx.x < 32) {
        float v = (lane < (kThreads / 32)) ? lds[lane] : 0.0f;
        #pragma unroll
        for (int o = 4; o > 0; o >>= 1) v += __shfl_down(v, o, 32);
        if (lane == 0) partial[blockIdx.x] = v;
    }
}

// Stage 2: single block folds the 2048 partials deterministically.
// Final 32-lane fold goes through V_WMMA_F32_16X16X4_F32 (A = lane values,
// B = ones -> each D column = sum of A rows; summing the 8 D VGPRs on lane 0
// yields the total in fp32).
__global__ void huber_final_kernel(const float* __restrict__ partial,
                                   float* __restrict__ out,
                                   int nparts, float inv_n) {
    float s = 0.0f;
    for (int i = threadIdx.x; i < nparts; i += blockDim.x) s += partial[i];
    s = wave32_reduce(s);

    __shared__ float lds[kThreads / 32];
    const int lane = threadIdx.x & 31;
    const int wid  = threadIdx.x >> 5;
    if (lane == 0) lds[wid] = s;
    __syncthreads();

    if (threadIdx.x < 32) {           // wave 0 only; EXEC all-1s inside
        float v = (lane < (kThreads / 32)) ? lds[lane] : 0.0f;
#if __has_builtin(__builtin_amdgcn_wmma_f32_16x16x4_f32)
        // A (16x4 f32): VGPR0 = per-lane value, VGPR1 = 0.
        //   lanes 0-15 feed A[m][0], lanes 16-31 feed A[m][2]; K=1,3 are zero.
        // B (4x16) = all ones regardless of striping.
        // D[m][n] = v(m) + v(m+16); sum of the 8 D VGPRs on lane 0 =
        //   sum_{m=0..7} (v(m)+v(m+16)); lanes 8..15 folded via lane16 rows = 0 here
        //   since only lanes 0..7 carry wave sums (kThreads/32 == 8).
        v2f a; a[0] = v;    a[1] = 0.0f;
        v2f b; b[0] = 1.0f; b[1] = 1.0f;
        v8f c = {};
        c = __builtin_amdgcn_wmma_f32_16x16x4_f32(
                /*neg_a=*/false, a, /*neg_b=*/false, b,
                /*c_mod=*/(short)0, c, /*reuse_a=*/false, /*reuse_b=*/false);
        float tot = c[0] + c[1] + c[2] + c[3] + c[4] + c[5] + c[6] + c[7];
#else
        #pragma unroll
        for (int o = 4; o > 0; o >>= 1) v += __shfl_down(v, o, 32);
        float tot = v;
#endif
        if (threadIdx.x == 0) out[0] = tot * inv_n;
    }
}

extern "C" void kernel_launch(void* const* d_in, const int* in_sizes, int n_in,
                              void* d_out, int out_size, void* d_ws, size_t ws_size,
                              hipStream_t stream) {
    const float* x = (const float*)d_in[0];
    const float* y = (const float*)d_in[1];
    float* out     = (float*)d_out;
    float* partial = (float*)d_ws;      // kBlocks floats = 8 KB scratch
    const int n = in_sizes[0];

    huber_partial_kernel<<<kBlocks, kThreads, 0, stream>>>(x, y, partial, n);
    huber_final_kernel<<<1, kThreads, 0, stream>>>(partial, out, kBlocks,
                                                   1.0f / (float)n);
}